// MonarchMatrix_4286377362096
// MI455X (gfx1250) — compile-verified
//
#include <hip/hip_runtime.h>
#include <hip/hip_bf16.h>

typedef float v2f __attribute__((ext_vector_type(2)));
typedef float v8f __attribute__((ext_vector_type(8)));

#define NCOL     4096
#define MT       16            // rows per workgroup (one WMMA M tile)
#define THREADS  512           // 16 waves (wave32) -> ~256 VGPR/lane headroom
#define XT_TS    273           // t-stride (dwords) in Xt chunk buffer (odd -> bank-clean)
#define XT_RS    17            // r-stride (dwords) inside a t slab (16 i's + 1 pad)
#define XT_BUF   (64 * XT_TS)  // 17472 dwords per chunk buffer
#define ZT_CS    1028          // c-stride (dwords) in Zt (16B-aligned spill, bank-clean reads)
#define SMEM_DW  (64 * ZT_CS)  // 65792 dwords = 263168 bytes (Zt; Xt buffers overlay it)

__global__ __launch_bounds__(THREADS, 1)
void monarch_fused_kernel(const float* __restrict__ x,
                          const float* __restrict__ L,
                          const float* __restrict__ R,
                          float* __restrict__ out)
{
    __shared__ float smem[SMEM_DW];

    const int tid  = threadIdx.x;
    const int lane = tid & 31;
    const int wave = tid >> 5;        // 0..15
    const int lm   = lane & 15;       // WMMA M/N lane index
    const int lh   = lane >> 4;       // 0/1 -> K offset 2*lh, M offset 8*lh for C/D
    const size_t rowbase = (size_t)blockIdx.x * MT;
    const float* xblk = x + rowbase * NCOL;

    // ---- async transposed chunk loader: chunk ch covers i in [16*ch, 16*ch+16) ----
    auto load_chunk = [&](int ch, float* buf) {
        #pragma unroll
        for (int q = 0; q < 32; ++q) {
            int idx = q * THREADS + tid;
            int t   = idx & 63;
            int il  = (idx >> 6) & 15;
            int r   = idx >> 10;
            unsigned long long ga =
                (unsigned long long)(xblk + (size_t)r * NCOL + (ch * 16 + il) * 64 + t);
            unsigned lo = (unsigned)(size_t)(buf + t * XT_TS + r * XT_RS + il);
            asm volatile("global_load_async_to_lds_b32 %0, %1, off"
                         :: "v"(lo), "v"(ga) : "memory");
        }
    };

    // ---- L-fragment group loader: 16 x global_load_b64, issued as one batch ----
    auto loadB1 = [&](int ch, int kk, v2f (&bf)[4][4]) {
        const int mg = ch * 16 + kk * 4 + 2 * lh;         // even -> 8B aligned
        #pragma unroll
        for (int bb = 0; bb < 4; ++bb) {
            const int b = wave * 4 + bb;
            #pragma unroll
            for (int ct = 0; ct < 4; ++ct)
                bf[bb][ct] = *(const v2f*)(L + (size_t)b * 4096 + (ct * 16 + lm) * 64 + mg);
        }
    };

    // ================= Stage 1: Z[b][c] = sum_m L[b][c][m] * x[m*64+b] =================
    v8f acc[4][4];
    #pragma unroll
    for (int bb = 0; bb < 4; ++bb)
        #pragma unroll
        for (int ct = 0; ct < 4; ++ct)
            acc[bb][ct] = v8f{0.f, 0.f, 0.f, 0.f, 0.f, 0.f, 0.f, 0.f};

    load_chunk(0, smem);                                  // prologue
    asm volatile("s_wait_asynccnt 0x0" ::: "memory");
    __syncthreads();

    #pragma unroll
    for (int ch = 0; ch < 4; ++ch) {
        float* cur = smem + (ch & 1) * XT_BUF;
        if (ch < 3) load_chunk(ch + 1, smem + ((ch + 1) & 1) * XT_BUF);

        v2f bf[4][4], bfn[4][4];
        loadB1(ch, 0, bf);                                // pipeline prologue
        #pragma unroll
        for (int kk = 0; kk < 4; ++kk) {                  // 4 K-steps (K=4 each) per chunk
            if (kk < 3) loadB1(ch, kk + 1, bfn);          // prefetch next K-step's L frags

            v2f a[4];
            #pragma unroll
            for (int bb = 0; bb < 4; ++bb) {
                const float* ap = cur + (wave * 4 + bb) * XT_TS + lm * XT_RS + kk * 4 + 2 * lh;
                a[bb].x = ap[0]; a[bb].y = ap[1];         // ds_load_2addr_b32
            }
            #pragma unroll
            for (int bb = 0; bb < 4; ++bb)
                #pragma unroll
                for (int ct = 0; ct < 4; ++ct)
                    acc[bb][ct] = __builtin_amdgcn_wmma_f32_16x16x4_f32(
                        false, a[bb], false, bf[bb][ct], (short)0, acc[bb][ct], false, false);

            #pragma unroll
            for (int bb = 0; bb < 4; ++bb)
                #pragma unroll
                for (int ct = 0; ct < 4; ++ct)
                    bf[bb][ct] = bfn[bb][ct];             // renamed away by unrolling
        }
        asm volatile("s_wait_asynccnt 0x0" ::: "memory");
        __syncthreads();   // chunk ch+1 visible; buffer (ch-1) readers long done
    }

    // ---- spill accumulators to transposed Zt[c][j][r] via ds_store_b128 ----
    #pragma unroll
    for (int bb = 0; bb < 4; ++bb) {
        const int b = wave * 4 + bb;
        #pragma unroll
        for (int ct = 0; ct < 4; ++ct) {
            const int c = ct * 16 + lm;                   // D: N = lm
            float* zp = smem + (size_t)c * ZT_CS + b * 16 + 8 * lh;   // 16B aligned
            *(float4*)(zp + 0) = make_float4(acc[bb][ct][0], acc[bb][ct][1],
                                             acc[bb][ct][2], acc[bb][ct][3]);
            *(float4*)(zp + 4) = make_float4(acc[bb][ct][4], acc[bb][ct][5],
                                             acc[bb][ct][6], acc[bb][ct][7]);
        }
    }
    __syncthreads();

    // ================= Stage 2: out[k*64+c] = sum_j R[c][k][j] * Z[j][c] =================
    auto loadB2 = [&](int kj, v2f (&bf)[4][4]) {
        const int j0 = kj * 4 + 2 * lh;                   // even -> 8B aligned
        #pragma unroll
        for (int cc = 0; cc < 4; ++cc) {
            const int c = wave * 4 + cc;
            #pragma unroll
            for (int kt = 0; kt < 4; ++kt)
                bf[cc][kt] = *(const v2f*)(R + (size_t)c * 4096 + (kt * 16 + lm) * 64 + j0);
        }
    };

    v8f acc2[4][4];
    #pragma unroll
    for (int cc = 0; cc < 4; ++cc)
        #pragma unroll
        for (int kt = 0; kt < 4; ++kt)
            acc2[cc][kt] = v8f{0.f, 0.f, 0.f, 0.f, 0.f, 0.f, 0.f, 0.f};

    {
        v2f bf[4][4], bfn[4][4];
        loadB2(0, bf);                                    // pipeline prologue
        #pragma unroll
        for (int kj = 0; kj < 16; ++kj) {                 // 16 K-steps over j
            if (kj < 15) loadB2(kj + 1, bfn);             // prefetch next K-step's R frags

            const int j0 = kj * 4 + 2 * lh;
            v2f a[4];
            #pragma unroll
            for (int cc = 0; cc < 4; ++cc) {
                const float* zp = smem + (size_t)(wave * 4 + cc) * ZT_CS + j0 * 16 + lm;
                a[cc].x = zp[0]; a[cc].y = zp[16];        // bank-conflict-free ds loads
            }
            #pragma unroll
            for (int cc = 0; cc < 4; ++cc)
                #pragma unroll
                for (int kt = 0; kt < 4; ++kt)
                    acc2[cc][kt] = __builtin_amdgcn_wmma_f32_16x16x4_f32(
                        false, a[cc], false, bf[cc][kt], (short)0, acc2[cc][kt], false, false);

            #pragma unroll
            for (int cc = 0; cc < 4; ++cc)
                #pragma unroll
                for (int kt = 0; kt < 4; ++kt)
                    bf[cc][kt] = bfn[cc][kt];
        }
    }

    // ---- store: out[row, (kt*16+lm)*64 + c]; WG covers full cachelines -> GL2 merges ----
    #pragma unroll
    for (int cc = 0; cc < 4; ++cc) {
        const int c = wave * 4 + cc;
        #pragma unroll
        for (int kt = 0; kt < 4; ++kt) {
            #pragma unroll
            for (int v = 0; v < 8; ++v) {
                const size_t row = rowbase + v + 8 * lh;
                out[row * NCOL + (size_t)(kt * 16 + lm) * 64 + c] = acc2[cc][kt][v];
            }
        }
    }
}

extern "C" void kernel_launch(void* const* d_in, const int* in_sizes, int n_in,
                              void* d_out, int out_size, void* d_ws, size_t ws_size,
                              hipStream_t stream) {
    const float* x = (const float*)d_in[0];
    const float* L = (const float*)d_in[1];
    const float* R = (const float*)d_in[2];
    float* out = (float*)d_out;

    const int nrows   = in_sizes[0] / NCOL;  // 16384 for the reference shapes
    const int nblocks = nrows / MT;          // 1024 workgroups of 16 rows

    monarch_fused_kernel<<<dim3(nblocks), dim3(THREADS), 0, stream>>>(x, L, R, out);
}